// NaturalVariation_61821759259047
// MI455X (gfx1250) — compile-verified
//
#include <hip/hip_runtime.h>
#include <math.h>

typedef float v2f __attribute__((ext_vector_type(2)));
typedef float v4f __attribute__((ext_vector_type(4)));
typedef float v8f __attribute__((ext_vector_type(8)));

#define B_  4
#define T_  8192
#define H_  1024
#define P_  8
#define NPAD 16              // B-matrix columns padded to WMMA N=16 with zeros
#define LC  128              // chunk length along T
#define CC  (T_ / LC)        // 64 chunks
#define SMOOTH      0.9f
#define EPS_COEF    (0.1f * 0.05f)   // (1-s) * VARIATION_SCALE*TEMPERATURE
#define WT_STRIDE   1032     // padded LDS stride for transposed W (kills bank conflicts)

// ---------------------------------------------------------------------------
// Kernel 1: chunk-local EMA endpoints (zero initial state per chunk).
// Grid: (CC, B), block 256. Thread owns 4 consecutive h (float4 streaming).
// ---------------------------------------------------------------------------
__global__ void __launch_bounds__(256) ema_chunk_local(
    const float* __restrict__ noise, float* __restrict__ carry) {
  const int c = blockIdx.x, b = blockIdx.y;
  const int h = threadIdx.x * 4;
  v4f s = {0.f, 0.f, 0.f, 0.f};
  const float* np = noise + ((size_t)b * T_ + (size_t)c * LC) * H_ + h;
  for (int t = 0; t < LC; ++t) {
    v4f e = *(const v4f*)np;
    s = SMOOTH * s + EPS_COEF * e;
    np += H_;
  }
  *(v4f*)(carry + (size_t)(b * CC + c) * H_ + h) = s;
}

// ---------------------------------------------------------------------------
// Kernel 2: sequential prefix across the 64 chunk endpoints -> per-chunk
// initial states (in place). 1024 threads total, each owns (b, h-quad).
// ---------------------------------------------------------------------------
__global__ void __launch_bounds__(256) ema_chunk_prefix(float* __restrict__ carry) {
  const int idx = blockIdx.x * blockDim.x + threadIdx.x;   // 0 .. B*H/4-1
  const int b = idx / (H_ / 4);
  const int h = (idx % (H_ / 4)) * 4;
  const float decay = __powf(SMOOTH, (float)LC);           // 0.9^128
  v4f run = {0.f, 0.f, 0.f, 0.f};
  for (int c = 0; c < CC; ++c) {
    float* p = carry + (size_t)(b * CC + c) * H_ + h;
    v4f local = *(v4f*)p;
    *(v4f*)p = run;                 // initial state entering chunk c
    run = decay * run + local;      // full-history state at end of chunk c
  }
}

// ---------------------------------------------------------------------------
// Kernel 3: fused main pass. One workgroup per (chunk, batch): 128 tokens x H.
// Phase A: logits GEMM via V_WMMA_F32_16X16X4_F32 (wave = 16-token tile).
// Phase B: softmax-weighted rank-8 variation + EMA + residual, streamed.
// ---------------------------------------------------------------------------
__global__ void __launch_bounds__(256) fused_main(
    const float* __restrict__ x, const float* __restrict__ Wsel,
    const float* __restrict__ bsel, const float* __restrict__ patterns,
    const float* __restrict__ noise, const float* __restrict__ carry,
    float* __restrict__ out) {
  __shared__ float sWt[NPAD * WT_STRIDE];  // W_sel^T [p][h]; rows 8..15 = zeros
  __shared__ float sLog[LC * P_];
  __shared__ float sWgt[LC * P_];

  const int c = blockIdx.x, b = blockIdx.y;
  const int tid = threadIdx.x;

  // Stage W_sel transposed into LDS rows 0..7 (coalesced global read),
  // and zero-fill rows 8..15 so B fragments need no runtime masking.
  for (int i = tid; i < H_ * P_; i += 256) {
    const int h = i >> 3, p = i & 7;
    sWt[p * WT_STRIDE + h] = Wsel[i];
  }
  for (int i = tid; i < P_ * WT_STRIDE; i += 256)
    sWt[P_ * WT_STRIDE + i] = 0.0f;
  __syncthreads();

  // ---------------- Phase A: logits via WMMA f32 16x16x4 ----------------
  // A tile: 16 tokens x 4 of H.  lane&15 = M (token) for A / N (pattern) for B;
  // half-wave selects K pair (0,1) vs (2,3) per ISA layout tables.
  const int wave = tid >> 5;
  const int lane = tid & 31;
  const int m    = lane & 15;
  const int kk   = (lane >> 4) << 1;              // 0 or 2
  const int tok0 = c * LC + wave * 16;

  const float* xa = x + ((size_t)b * T_ + tok0 + m) * H_ + kk;
  const float* wb = &sWt[m * WT_STRIDE + kk];     // rows 8..15 are zeros

  v8f acc = {};
  for (int h0 = 0; h0 < H_; h0 += 4) {
    v2f a  = *(const v2f*)(xa + h0);     // A fragment (K pair for this half-wave)
    v2f bb = *(const v2f*)(wb + h0);     // B fragment straight from LDS
    acc = __builtin_amdgcn_wmma_f32_16x16x4_f32(
        /*neg_a=*/false, a, /*neg_b=*/false, bb,
        /*c_mod=*/(short)0, acc, /*reuse_a=*/false, /*reuse_b=*/false);
  }

  // Scatter valid logits (pattern columns 0..7) to LDS.
  // C/D layout: VGPR g -> M=g (lanes 0-15), M=g+8 (lanes 16-31), N=lane&15.
  if (m < P_) {
    const int tl_base = wave * 16 + ((lane < 16) ? 0 : 8);
#pragma unroll
    for (int g = 0; g < 8; ++g)
      sLog[(tl_base + g) * P_ + m] = acc[g];
  }
  __syncthreads();

  // Softmax over 8 patterns, one token per thread (threads 0..127).
  if (tid < LC) {
    float l[P_];
    float mx = -3.4e38f;
#pragma unroll
    for (int p = 0; p < P_; ++p) {
      l[p] = sLog[tid * P_ + p] + bsel[p];
      mx = fmaxf(mx, l[p]);
    }
    float sum = 0.f;
#pragma unroll
    for (int p = 0; p < P_; ++p) { l[p] = __expf(l[p] - mx); sum += l[p]; }
    const float inv = 1.0f / sum;
#pragma unroll
    for (int p = 0; p < P_; ++p) sWgt[tid * P_ + p] = l[p] * inv;
  }
  __syncthreads();

  // ---------------- Phase B: variation + EMA + residual ----------------
  const int h = tid * 4;
  v4f pat[P_];
#pragma unroll
  for (int p = 0; p < P_; ++p)
    pat[p] = *(const v4f*)(patterns + p * H_ + h);

  v4f state = *(const v4f*)(carry + (size_t)(b * CC + c) * H_ + h);

  const size_t base = ((size_t)b * T_ + (size_t)c * LC) * H_ + h;
  const float* np = noise + base;
  const float* xp = x + base;
  float*       op = out + base;

  for (int tl = 0; tl < LC; ++tl) {
    v4f e  = *(const v4f*)np;
    state  = SMOOTH * state + EPS_COEF * e;
    v4f xv = *(const v4f*)xp;
    v4f var = {0.f, 0.f, 0.f, 0.f};
#pragma unroll
    for (int p = 0; p < P_; ++p)
      var += sWgt[tl * P_ + p] * pat[p];   // LDS broadcast reads
    *(v4f*)op = xv + var + state;
    np += H_; xp += H_; op += H_;
  }
}

// ---------------------------------------------------------------------------
extern "C" void kernel_launch(void* const* d_in, const int* in_sizes, int n_in,
                              void* d_out, int out_size, void* d_ws, size_t ws_size,
                              hipStream_t stream) {
  const float* x        = (const float*)d_in[0];
  const float* Wsel     = (const float*)d_in[1];
  const float* bsel     = (const float*)d_in[2];
  const float* patterns = (const float*)d_in[3];
  const float* noise    = (const float*)d_in[4];
  float*       out      = (float*)d_out;
  float*       carry    = (float*)d_ws;   // B*CC*H floats = 1 MB

  ema_chunk_local<<<dim3(CC, B_), 256, 0, stream>>>(noise, carry);
  ema_chunk_prefix<<<dim3((B_ * H_ / 4) / 256), 256, 0, stream>>>(carry);
  fused_main<<<dim3(CC, B_), 256, 0, stream>>>(x, Wsel, bsel, patterns,
                                               noise, carry, out);
}